// ElasticARC_69295002354040
// MI455X (gfx1250) — compile-verified
//
#include <hip/hip_runtime.h>

// ElasticArcFace margin head for MI455X (gfx1250).
//
// out[i,j] = 64 * logits[i,j]                                  (all j != label[i])
// out[i,L] = 64 * (x*cos(m_i) - sqrt(1-x^2)*sin(m_i)), x = logits[i,L]
//
// Memory-bound: 1.64 GB streamed once -> NT-hinted b128 loads/stores.

typedef float v4f __attribute__((ext_vector_type(4)));

#define S_SCALE 64.0f

__global__ __launch_bounds__(256)
void elastic_arc_scale(const v4f* __restrict__ in, v4f* __restrict__ out, long long n4) {
    long long idx    = (long long)blockIdx.x * blockDim.x + threadIdx.x;
    long long stride = (long long)gridDim.x * blockDim.x;
    for (long long i = idx; i < n4; i += stride) {
        // Single-touch stream: non-temporal on both sides (TH=NT), keeps the
        // 192 MB L2 from being churned by 819 MB of one-shot data.
        v4f v = __builtin_nontemporal_load(in + i);
        v *= S_SCALE;
        __builtin_nontemporal_store(v, out + i);
    }
}

__global__ __launch_bounds__(256)
void elastic_arc_fixup(const float* __restrict__ logits,
                       const int*   __restrict__ labels,
                       const float* __restrict__ elastic,
                       float*       __restrict__ out,
                       int nrows, long long ncols) {
    int i = blockIdx.x * blockDim.x + threadIdx.x;
    if (i >= nrows) return;
    long long lab = (long long)labels[i];
    if (lab < 0 || lab >= ncols) return;          // ignore-index rows: margin = 0
    long long off = (long long)i * ncols + lab;
    float x  = logits[off];
    float m  = elastic[i];
    float sm = __sinf(m);
    float cm = __cosf(m);
    // cos(arccos(x) + m) = x*cos(m) - sin(arccos(x))*sin(m); x in [0,1)
    float sint = sqrtf(fmaxf(0.0f, 1.0f - x * x));
    out[off] = S_SCALE * (x * cm - sint * sm);
}

extern "C" void kernel_launch(void* const* d_in, const int* in_sizes, int n_in,
                              void* d_out, int out_size, void* d_ws, size_t ws_size,
                              hipStream_t stream) {
    const float* logits  = (const float*)d_in[0];
    const int*   labels  = (const int*)  d_in[1];
    const float* elastic = (const float*)d_in[2];
    float*       out     = (float*)d_out;

    const int       nrows = in_sizes[1];              // B (2048)
    const long long total = (long long)out_size;      // B*C
    const long long ncols = total / nrows;            // C (100000, divisible by 4)
    const long long n4    = total >> 2;               // float4 count

    const int threads = 256;
    long long want = (n4 + threads - 1) / threads;
    int blocks = (int)(want > (1LL << 20) ? (1LL << 20) : want);
    elastic_arc_scale<<<blocks, threads, 0, stream>>>(
        (const v4f*)logits, (v4f*)out, n4);

    int fblocks = (nrows + threads - 1) / threads;
    elastic_arc_fixup<<<fblocks, threads, 0, stream>>>(
        logits, labels, elastic, out, nrows, ncols);
}